// DiscriminativeLoss_40157944217762
// MI455X (gfx1250) — compile-verified
//
#include <hip/hip_runtime.h>
#include <hip/hip_bf16.h>

typedef __attribute__((ext_vector_type(2))) float v2f;
typedef __attribute__((ext_vector_type(8))) float v8f;

#define F_CH   32
#define K_LBL  32
#define N_PIX  (512 * 512)
#define B_SZ   8
#define DELTA_V 0.5f
#define DELTA_D 1.5f
#define EPS_F   1e-12f

// CDNA5 async global->LDS staging (ASYNCcnt-tracked), guarded so the file
// still compiles on toolchains without the builtins.
#if defined(__has_builtin)
#  if __has_builtin(__builtin_amdgcn_global_load_async_to_lds_b32) && \
      __has_builtin(__builtin_amdgcn_s_wait_asynccnt)
#    define USE_ASYNC_LDS 1
#  endif
#endif
#ifndef USE_ASYNC_LDS
#  define USE_ASYNC_LDS 0
#endif

#if USE_ASYNC_LDS
#  define AS1 __attribute__((address_space(1)))
#  define AS3 __attribute__((address_space(3)))
#endif

struct Tile {
  float xt[32][257];   // +1 pad: LDS bank = (f + pix) mod 64 -> conflict-free
  int   lb[256];
};

__device__ __forceinline__ void wait_async_lds() {
#if USE_ASYNC_LDS
  __builtin_amdgcn_s_wait_asynccnt(0);
#endif
}

__device__ __forceinline__ void stage_tile(const float* __restrict__ xb,
                                           const int* __restrict__ lbp,
                                           int p, int t, Tile* dst, int* cnt) {
  #pragma unroll
  for (int f = 0; f < 32; ++f) {
#if USE_ASYNC_LDS
    __builtin_amdgcn_global_load_async_to_lds_b32(
        (AS1 int*)(xb + (size_t)f * N_PIX + p),
        (AS3 int*)&dst->xt[f][t], /*offset=*/0, /*cpol=*/0);
#else
    dst->xt[f][t] = xb[(size_t)f * N_PIX + p];   // coalesced per channel
#endif
  }
  const int l = lbp[p];
  dst->lb[t] = l;
  atomicAdd(&cnt[l], 1);
}

// ---------------------------------------------------------------------------
// Pass 1: sums[b,f,k] = sum_n x[b,f,n] * (label[b,n]==k), counts[b,k]
// WMMA f32 16x16x4:  A = x tile [16 ch x 4 pix],  B = one-hot [4 pix x 16 lbl]
// Per wave: persistent 2x2 accumulator tiles covering F=32 x K=32.
// 64 blocks/batch, double-buffered LDS tiles (async copy overlaps WMMA).
// ---------------------------------------------------------------------------
__global__ __launch_bounds__(256) void k_sums(const float* __restrict__ x,
                                              const int* __restrict__ lab,
                                              float* __restrict__ gSums,
                                              float* __restrict__ gCounts) {
  __shared__ union {
    Tile  tile[2];
    float partial[8][1024];  // per-wave partials for cross-wave reduction
  } u;
  __shared__ int cnt[32];

  const int t     = threadIdx.x;
  const int b     = blockIdx.x >> 6;   // 64 blocks per batch
  const int chunk = blockIdx.x & 63;
  const int pixBase = chunk * (N_PIX / 64);
  const int TILES = (N_PIX / 64) / 256;  // 16

  if (t < 32) cnt[t] = 0;
  __syncthreads();

  const int lane  = t & 31;
  const int wave  = t >> 5;
  const int fRow  = lane & 15;   // A-matrix row (channel) within 16-tile
  const int hi    = lane >> 4;   // high half-wave -> K-reduction rows 2,3
  const int col   = fRow;        // B/D column (label) within 16-tile
  const int wbase = wave * 32;   // this wave's 32 pixels inside the 256-tile

  v8f c00 = {}; v8f c01 = {}; v8f c10 = {}; v8f c11 = {};

  const float* xb  = x   + (size_t)b * F_CH * N_PIX;
  const int*   lbp = lab + (size_t)b * N_PIX;

  // prologue: fill buffer 0
  stage_tile(xb, lbp, pixBase + t, t, &u.tile[0], cnt);
  wait_async_lds();
  __syncthreads();

  for (int tile = 0; tile < TILES; ++tile) {
    const int cur = tile & 1;
    if (tile + 1 < TILES)   // prefetch next tile into the back buffer
      stage_tile(xb, lbp, pixBase + (tile + 1) * 256 + t, t, &u.tile[cur ^ 1], cnt);

    const Tile& tb = u.tile[cur];
    // WMMA: 8 groups of 4 pixels {p0,p0+1,p0+16,p0+17} -> conflict-free banks
    #pragma unroll
    for (int j = 0; j < 8; ++j) {
      const int p0 = wbase + j * 2 + hi * 16;
      const int p1 = p0 + 1;
      v2f A0 = { tb.xt[fRow     ][p0], tb.xt[fRow     ][p1] };  // ch 0..15
      v2f A1 = { tb.xt[fRow + 16][p0], tb.xt[fRow + 16][p1] };  // ch 16..31
      const int l0 = tb.lb[p0], l1 = tb.lb[p1];
      v2f B0 = { (l0 == col     ) ? 1.0f : 0.0f,
                 (l1 == col     ) ? 1.0f : 0.0f };              // lbl 0..15
      v2f B1 = { (l0 == col + 16) ? 1.0f : 0.0f,
                 (l1 == col + 16) ? 1.0f : 0.0f };              // lbl 16..31
      c00 = __builtin_amdgcn_wmma_f32_16x16x4_f32(false, A0, false, B0, (short)0, c00, false, false);
      c01 = __builtin_amdgcn_wmma_f32_16x16x4_f32(false, A0, false, B1, (short)0, c01, false, false);
      c10 = __builtin_amdgcn_wmma_f32_16x16x4_f32(false, A1, false, B0, (short)0, c10, false, false);
      c11 = __builtin_amdgcn_wmma_f32_16x16x4_f32(false, A1, false, B1, (short)0, c11, false, false);
    }
    wait_async_lds();   // back buffer fully resident before it becomes front
    __syncthreads();
  }

  // Cross-wave reduction in LDS, then one atomic per (f,k) per block.
  #pragma unroll
  for (int r = 0; r < 8; ++r) {         // D layout: VGPR r, lanes>=16 -> M=r+8
    const int f0 = r + hi * 8;
    u.partial[wave][ f0       * 32 + col     ] = c00[r];
    u.partial[wave][ f0       * 32 + col + 16] = c01[r];
    u.partial[wave][(f0 + 16) * 32 + col     ] = c10[r];
    u.partial[wave][(f0 + 16) * 32 + col + 16] = c11[r];
  }
  __syncthreads();
  float* gs = gSums + (size_t)b * 1024;
  #pragma unroll
  for (int i = 0; i < 4; ++i) {
    const int idx = t + i * 256;
    float s = 0.f;
    #pragma unroll
    for (int w = 0; w < 8; ++w) s += u.partial[w][idx];
    atomicAdd(&gs[idx], s);
  }
  if (t < 32) atomicAdd(&gCounts[b * 32 + t], (float)cnt[t]);
}

// ---------------------------------------------------------------------------
// Pass 2 (tiny): per batch -> mu, ||mu||^2, pixel weights, push + reg terms.
// ---------------------------------------------------------------------------
__global__ __launch_bounds__(256) void k_stats(const float* __restrict__ gSums,
                                               const float* __restrict__ gCounts,
                                               float* __restrict__ gMu,
                                               float* __restrict__ gMun2,
                                               float* __restrict__ gWlab,
                                               float* __restrict__ out) {
  __shared__ float mu[32][33];
  __shared__ float cnts[32];
  __shared__ float mun2s[32];
  __shared__ int   pres[32];
  __shared__ float Csh;
  __shared__ float red[256];

  const int t = threadIdx.x;
  const int b = blockIdx.x;

  if (t < 32) {
    const float c = gCounts[b * 32 + t];
    cnts[t] = c;
    pres[t] = (c > 0.f) ? 1 : 0;
  }
  __syncthreads();
  if (t == 0) {
    float C = 0.f;
    for (int k = 0; k < 32; ++k) C += pres[k] ? 1.f : 0.f;
    Csh = C;
  }
  #pragma unroll
  for (int i = 0; i < 4; ++i) {
    const int idx = t + i * 256;
    const int f = idx >> 5, k = idx & 31;
    const float m = gSums[b * 1024 + idx] / fmaxf(cnts[k], 1.f);
    mu[f][k] = m;
    gMu[b * 1024 + idx] = m;
  }
  __syncthreads();
  if (t < 32) {
    float s = 0.f;
    #pragma unroll
    for (int f = 0; f < 32; ++f) s += mu[f][t] * mu[f][t];
    mun2s[t] = s;
    gMun2[b * 32 + t] = s;
    gWlab[b * 32 + t] = pres[t] ? (1.0f / (cnts[t] * fmaxf(Csh, 1.f))) : 0.f;
  }
  __syncthreads();

  // push term: upper-triangle pairwise hinge between present cluster means
  float accD = 0.f;
  for (int idx = t; idx < 1024; idx += 256) {
    const int ca = idx >> 5, cb = idx & 31;
    if (ca < cb && pres[ca] && pres[cb]) {
      float s = 0.f;
      #pragma unroll
      for (int f = 0; f < 32; ++f) {
        const float d = mu[f][ca] - mu[f][cb];
        s += d * d;
      }
      const float dist = sqrtf(s + EPS_F);
      const float h = fmaxf(DELTA_D - dist, 0.f);
      accD += h * h;
    }
  }
  const float accR = (t < 32 && pres[t]) ? sqrtf(mun2s[t] + EPS_F) : 0.f;

  red[t] = accD;
  __syncthreads();
  for (int s = 128; s > 0; s >>= 1) { if (t < s) red[t] += red[t + s]; __syncthreads(); }
  const float dloss = red[0];
  __syncthreads();
  red[t] = accR;
  __syncthreads();
  for (int s = 128; s > 0; s >>= 1) { if (t < s) red[t] += red[t + s]; __syncthreads(); }
  if (t == 0) {
    const float C = Csh;
    const float disB = (C > 2.f) ? dloss / fmaxf(C * (C - 1.f), 1.f) : 0.f;
    const float regB = (C > 1.f) ? red[0] : 0.f;
    atomicAdd(out, /*BETA*/ 1.0f * disB + /*GAMMA*/ 0.001f * regB);
  }
}

// ---------------------------------------------------------------------------
// Pass 3: pull term. Second (and last) HBM stream of embeds.
// d^2 = ||x||^2 - 2 x.mu[lab] + ||mu[lab]||^2 ; hinge^2 weighted by 1/(cnt*C).
// ---------------------------------------------------------------------------
__global__ __launch_bounds__(256) void k_var(const float* __restrict__ x,
                                             const int* __restrict__ lab,
                                             const float* __restrict__ gMu,
                                             const float* __restrict__ gMun2,
                                             const float* __restrict__ gWlab,
                                             float* __restrict__ out) {
  __shared__ float mu[32][33];
  __shared__ float mun2s[32];
  __shared__ float wl[32];
  __shared__ float red[256];

  const int t     = threadIdx.x;
  const int b     = blockIdx.x >> 7;   // 128 blocks per batch
  const int chunk = blockIdx.x & 127;
  const int pixBase = chunk * (N_PIX / 128);

  #pragma unroll
  for (int i = 0; i < 4; ++i) {
    const int idx = t + i * 256;
    mu[idx >> 5][idx & 31] = gMu[b * 1024 + idx];
  }
  if (t < 32) {
    mun2s[t] = gMun2[b * 32 + t];
    wl[t]    = gWlab[b * 32 + t];
  }
  __syncthreads();

  const float* xb  = x   + (size_t)b * F_CH * N_PIX;
  const int*   lbp = lab + (size_t)b * N_PIX;

  float acc = 0.f;
  for (int it = 0; it < (N_PIX / 128) / 256; ++it) {   // 8 iterations
    const int p = pixBase + it * 256 + t;
    const int l = lbp[p];
    float dot = 0.f, x2 = 0.f;
    #pragma unroll
    for (int f = 0; f < 32; ++f) {
      const float v = xb[(size_t)f * N_PIX + p];       // coalesced per channel
      dot = fmaf(v, mu[f][l], dot);
      x2  = fmaf(v, v, x2);
    }
    const float s = fmaxf(x2 - 2.f * dot + mun2s[l], 0.f);
    const float d = sqrtf(s + EPS_F);
    const float h = fmaxf(d - DELTA_V, 0.f);
    acc += h * h * wl[l];
  }

  red[t] = acc;
  __syncthreads();
  for (int s = 128; s > 0; s >>= 1) { if (t < s) red[t] += red[t + s]; __syncthreads(); }
  if (t == 0) atomicAdd(out, red[0]);   // ALPHA = 1
}

// ---------------------------------------------------------------------------
extern "C" void kernel_launch(void* const* d_in, const int* in_sizes, int n_in,
                              void* d_out, int out_size, void* d_ws, size_t ws_size,
                              hipStream_t stream) {
  const float* embeds = (const float*)d_in[0];
  const int*   labels = (const int*)d_in[1];

  float* ws      = (float*)d_ws;
  float* gSums   = ws;                 // B*1024
  float* gCounts = ws + 8192;          // B*32
  float* gMu     = ws + 8448;          // B*1024
  float* gMun2   = ws + 16640;         // B*32
  float* gWlab   = ws + 16896;         // B*32

  (void)hipMemsetAsync(d_ws, 0, (size_t)(8192 + 256) * sizeof(float), stream);
  (void)hipMemsetAsync(d_out, 0, sizeof(float), stream);

  k_sums <<<dim3(B_SZ * 64),  dim3(256), 0, stream>>>(embeds, labels, gSums, gCounts);
  k_stats<<<dim3(B_SZ),       dim3(256), 0, stream>>>(gSums, gCounts, gMu, gMun2, gWlab, (float*)d_out);
  k_var  <<<dim3(B_SZ * 128), dim3(256), 0, stream>>>(embeds, labels, gMu, gMun2, gWlab, (float*)d_out);
}